// ConvLayer_49357764165670
// MI455X (gfx1250) — compile-verified
//
#include <hip/hip_runtime.h>

// Problem constants (match reference)
#define NB   4096   // batch (samples)
#define NF   200    // feature rows per sample
#define NK   40     // K (num_features)
#define NC   128    // out channels

typedef __attribute__((ext_vector_type(2))) float v2f;
typedef __attribute__((ext_vector_type(4))) float v4f;
typedef __attribute__((ext_vector_type(8))) float v8f;

// Row-groups of 16 rows covering NF=200 rows: ceil(200/16) = 13
#define NRG  13

__device__ __forceinline__ void sched_fence() {
#if defined(__has_builtin)
#if __has_builtin(__builtin_amdgcn_sched_barrier)
    __builtin_amdgcn_sched_barrier(0);
#endif
#endif
}

// Pipeline group pin: best-effort ordered instruction groups for the scheduler.
// Template non-type params keep the builtin's args integer-constant expressions.
template <int MASK, int SIZE>
__device__ __forceinline__ void sched_group() {
#if defined(__has_builtin)
#if __has_builtin(__builtin_amdgcn_sched_group_barrier)
    __builtin_amdgcn_sched_group_barrier(MASK, SIZE, 0);
#endif
#endif
}
#define SG_DS_READ 0x100
#define SG_WMMA    0x008

__global__ __launch_bounds__(256)
void softmax_gemm_reduce_kernel(const float* __restrict__ x,
                                const float* __restrict__ W,
                                float* __restrict__ out)
{
    // W staged in LDS, pair-interleaved: element (k, c) lives at
    //   sW[(k>>1)*256 + c*2 + (k&1)]
    // so a WMMA B-fragment (two consecutive K values at one column) is a
    // single aligned 8-byte ds_load (pairs merge into ds_load_2addr_b64).
    __shared__ float sW[(NK / 2) * (NC * 2)];   // 5120 floats (20 KB)
    __shared__ float sOut[NC];                  // per-sample output accumulator

    const int tid  = threadIdx.x;
    const int b    = blockIdx.x;
    const int wave = tid >> 5;   // 0..7
    const int lane = tid & 31;
    const int half = lane >> 4;  // 0: lanes 0-15, 1: lanes 16-31
    const int lm   = lane & 15;

    // ---- stage W into LDS (vectorized read, interleaved write) ----
    for (int i4 = tid; i4 < (NK * NC) / 4; i4 += 256) {
        const int i = i4 * 4;                  // 4 consecutive c, same k row
        const int k = i >> 7;
        const int c = i & (NC - 1);
        v4f w = *(const v4f*)(W + i);
        float* dst = &sW[(k >> 1) * 256 + c * 2 + (k & 1)];
        dst[0] = w.x; dst[2] = w.y; dst[4] = w.z; dst[6] = w.w;
    }
    if (tid < NC) sOut[tid] = 0.0f;
    __syncthreads();

    // Base of this lane's B-fragment column in the interleaved W image.
    const float* sWcol = &sW[lm * 2];

    // ---- each wave processes 16-row tiles of this sample ----
    for (int rg = wave; rg < NRG; rg += 8) {
        // A fragments: lane holds row m = lm; half selects the K parity pair.
        // Per ISA layout for V_WMMA_F32_16X16X4_F32 A (16x4):
        //   lanes 0-15:  K = 4k+0, 4k+1 ; lanes 16-31: K = 4k+2, 4k+3
        const int f       = rg * 16 + lm;
        const int fclamp  = (f < NF) ? f : (NF - 1);   // clamp OOB row (masked later)
        const float* xrow = x + ((size_t)b * NF + fclamp) * NK;
        const int klo     = half * 2;

        v2f afr[10];
        #pragma unroll
        for (int ks = 0; ks < 10; ++ks)
            afr[ks] = *(const v2f*)(xrow + 4 * ks + klo);   // 8B aligned, one clause

        // B double buffer: prologue-load ks=0 fragments alongside the A clause
        // so VMEM and DS latency overlap.
        v2f bf[2][8];
        #pragma unroll
        for (int nt = 0; nt < 8; ++nt)
            bf[0][nt] = *(const v2f*)(sWcol + (half)*256 + nt * 32);

        // Keep all prologue loads issued before any WMMA (partial waits only).
        sched_fence();

        // 16x128 logits in registers: 8 column tiles x v8f accumulators.
        // Pinned pipeline: [DS x4 (ks+1 B-frags)] ahead of [WMMA x8 (ks)].
        v8f acc[8] = {};
        #pragma unroll
        for (int ks = 0; ks < 10; ++ks) {
            const int cur = ks & 1;
            const int nxt = cur ^ 1;
            if (ks < 9) {
                const int j = 2 * (ks + 1) + half;   // next K-pair for this half
                #pragma unroll
                for (int nt = 0; nt < 8; ++nt)
                    bf[nxt][nt] = *(const v2f*)(sWcol + j * 256 + nt * 32);
                // 8 v2f loads merge into 4 ds_load_2addr_b64
                sched_group<SG_DS_READ, 4>();
            }
            #pragma unroll
            for (int nt = 0; nt < 8; ++nt) {
                acc[nt] = __builtin_amdgcn_wmma_f32_16x16x4_f32(
                              /*neg_a=*/false, afr[ks], /*neg_b=*/false, bf[cur][nt],
                              /*c_mod=*/(short)0, acc[nt],
                              /*reuse_a=*/false, /*reuse_b=*/false);
            }
            sched_group<SG_WMMA, 8>();
        }

        // ---- softmax per row (rows 0-7 in half 0, rows 8-15 in half 1) ----
        float rinv[8];
        #pragma unroll
        for (int r = 0; r < 8; ++r) {
            // Row max over 128 columns: in-lane over 8 tiles, then butterfly
            // over the 16-lane half (xor masks 1,2,4,8 stay within the half).
            float m = acc[0][r];
            #pragma unroll
            for (int nt = 1; nt < 8; ++nt) m = fmaxf(m, acc[nt][r]);
            #pragma unroll
            for (int s = 1; s < 16; s <<= 1) m = fmaxf(m, __shfl_xor(m, s, 32));

            float sum = 0.0f;
            #pragma unroll
            for (int nt = 0; nt < 8; ++nt) {
                float e = __expf(acc[nt][r] - m);
                acc[nt][r] = e;
                sum += e;
            }
            #pragma unroll
            for (int s = 1; s < 16; s <<= 1) sum += __shfl_xor(sum, s, 32);
            rinv[r] = 1.0f / sum;
        }

        // ---- normalize, mask invalid rows, reduce 16 rows -> column sums ----
        #pragma unroll
        for (int nt = 0; nt < 8; ++nt) {
            float cs = 0.0f;
            #pragma unroll
            for (int r = 0; r < 8; ++r) {
                const int frow = rg * 16 + half * 8 + r;   // C/D row = half*8 + r
                float v = acc[nt][r] * rinv[r];
                cs += (frow < NF) ? v : 0.0f;
            }
            cs += __shfl_xor(cs, 16, 32);   // merge the two lane halves
            if (half == 0) atomicAdd(&sOut[nt * 16 + lm], cs);  // ds_add_f32
        }
    }

    __syncthreads();
    if (tid < NC) out[(size_t)b * NC + tid] = sOut[tid];
}

extern "C" void kernel_launch(void* const* d_in, const int* in_sizes, int n_in,
                              void* d_out, int out_size, void* d_ws, size_t ws_size,
                              hipStream_t stream) {
    const float* x = (const float*)d_in[0];   // [B, F, N] f32
    const float* W = (const float*)d_in[1];   // [N, C]    f32
    float* out     = (float*)d_out;           // [B, C]    f32

    dim3 grid(NB);     // one block per sample
    dim3 block(256);   // 8 waves (wave32)
    softmax_gemm_reduce_kernel<<<grid, block, 0, stream>>>(x, W, out);
}